// SimpleModel_2791728743146
// MI455X (gfx1250) — compile-verified
//
#include <hip/hip_runtime.h>

#define T_TILES  200000
#define DMODEL   64
#define NPIECES  1000000
#define NNATIONS 16
#define KAUG     20     // 16 nations + cnt*piece_fc_b + 2 terrain one-hots + (tile_fc_b+active)

typedef __attribute__((ext_vector_type(2))) float v2f;
typedef __attribute__((ext_vector_type(8))) float v8f;

// ---------------- zero workspace (histogram + pooled accumulator) ----------------
__global__ void k_zero(unsigned int* __restrict__ natcnt, float* __restrict__ pooled) {
    int i = blockIdx.x * blockDim.x + threadIdx.x;
    if (i < T_TILES * NNATIONS) natcnt[i] = 0u;
    if (i < DMODEL) pooled[i] = 0.0f;
}

// ---------------- tiny projections: nation_proj[16x64], terrain_proj[2x64], active emb ----------------
__global__ void k_precompute(const float* __restrict__ nation_emb,
                             const float* __restrict__ terrain_emb,
                             const float* __restrict__ piece_fc_w,   // [128,64], top half used
                             const float* __restrict__ tile_fc_w,    // [128,64], bottom half used
                             const int*   __restrict__ active_nation,
                             float* __restrict__ natproj,
                             float* __restrict__ terrproj,
                             float* __restrict__ act) {
    const int tid = threadIdx.x;
    for (int idx = tid; idx < NNATIONS * DMODEL; idx += blockDim.x) {
        const int n = idx >> 6, d = idx & 63;
        float s = 0.0f;
        for (int k = 0; k < DMODEL; ++k)
            s += nation_emb[n * DMODEL + k] * piece_fc_w[k * DMODEL + d];
        natproj[idx] = s;
    }
    for (int idx = tid; idx < 2 * DMODEL; idx += blockDim.x) {
        const int e = idx >> 6, d = idx & 63;
        float s = 0.0f;
        for (int k = 0; k < DMODEL; ++k)
            s += terrain_emb[e * DMODEL + k] * tile_fc_w[(DMODEL + k) * DMODEL + d];
        terrproj[idx] = s;
    }
    if (tid < DMODEL) act[tid] = nation_emb[active_nation[0] * DMODEL + tid];
}

// ---------------- per-piece histogram: 4-byte atomics instead of 256-byte scatters ----------------
__global__ void k_hist(const int* __restrict__ nation_idxs,
                       const int* __restrict__ piece_tile_idxs,
                       unsigned int* __restrict__ natcnt) {
    int p = blockIdx.x * blockDim.x + threadIdx.x;
    if (p < NPIECES) {
        const int t = piece_tile_idxs[p];
        const int n = nation_idxs[p];
        atomicAdd(&natcnt[t * NNATIONS + n], 1u);
    }
}

// ---------------- main fused kernel: everything on the matrix pipe ----------------
// tile_state[t,d] = (tilepos@Wtop + Aaug@Baug)[t,d] + cnt_t * (ptile@Wbot)[t,d]
__global__ __launch_bounds__(256) void k_main(
    const int*   __restrict__ tile_idxs,
    const int*   __restrict__ terrain_types,
    const float* __restrict__ tile_pos_emb,     // [T,64]
    const float* __restrict__ piece_tile_emb,   // [T,64]
    const float* __restrict__ tile_fc_w,        // [128,64], rows 0..63 (top) used
    const float* __restrict__ tile_fc_b,        // [64]
    const float* __restrict__ piece_fc_w,       // [128,64], rows 64..127 (bottom) used
    const float* __restrict__ piece_fc_b,       // [64]
    const unsigned int* __restrict__ natcnt,    // [T,16]
    const float* __restrict__ natproj_g,        // [16,64]
    const float* __restrict__ terrproj_g,       // [2,64]
    const float* __restrict__ act_g,            // [64]
    const float* __restrict__ policy_w,         // [64]
    const float* __restrict__ policy_b,         // [1]
    float* __restrict__ out_logits,             // d_out[0..T-1]
    float* __restrict__ pooled_ws)              // [64] accumulator
{
    __shared__ __align__(16) float s_afrag[32 * KAUG];    // augmented A rows (stride 20, even -> b64 ok)
    __shared__ __align__(16) float s_bfragT[DMODEL * KAUG]; // augmented B, transposed per column d
    __shared__ float s_cnt[32];
    __shared__ float s_state[32 * 65];                    // pad row stride to 65 -> conflict-free

    const int tid    = threadIdx.x;
    const int block0 = blockIdx.x * 32;                   // first tile handled by this block

    // ---- stage augmented A fragments: one thread per row ----
    if (tid < 32) {
        const int r = tid;
        float c = 0.0f;
        #pragma unroll
        for (int n = 0; n < NNATIONS; ++n) {
            const float f = (float)natcnt[(size_t)(block0 + r) * NNATIONS + n];
            s_afrag[r * KAUG + n] = f;
            c += f;
        }
        s_afrag[r * KAUG + 16] = c;                       // scales piece_fc_b
        const int e = terrain_types[block0 + r] - 1;
        s_afrag[r * KAUG + 17] = (e == 0) ? 1.0f : 0.0f;  // terrain one-hot
        s_afrag[r * KAUG + 18] = (e == 1) ? 1.0f : 0.0f;
        s_afrag[r * KAUG + 19] = 1.0f;                    // picks up tile_fc_b + active emb
        s_cnt[r] = c;
    }
    // ---- stage augmented B (transposed): one thread per column d ----
    if (tid < DMODEL) {
        const int d = tid;
        #pragma unroll
        for (int n = 0; n < NNATIONS; ++n)
            s_bfragT[d * KAUG + n] = natproj_g[n * DMODEL + d];
        s_bfragT[d * KAUG + 16] = piece_fc_b[d];
        s_bfragT[d * KAUG + 17] = terrproj_g[d];
        s_bfragT[d * KAUG + 18] = terrproj_g[DMODEL + d];
        s_bfragT[d * KAUG + 19] = tile_fc_b[d] + act_g[d];
    }
    __syncthreads();

    // wave layout: 8 waves = 2 row-tiles x 4 col-tiles of the 32x64 output block
    const int wave  = tid >> 5;
    const int lane  = tid & 31;
    const int rtile = wave >> 2;            // 0..1  (16 rows each)
    const int n0    = (wave & 3) * 16;      // column tile base
    const int hl    = lane & 15;
    const int hi    = lane >> 4;            // K-half select per the 16x16x4 A/B layout

    const int  m_local = rtile * 16 + hl;                              // local A row (0..31)
    const long rowA    = (long)tile_idxs[block0 + m_local] * DMODEL;   // gather for tile_pos_emb
    const long rowB    = (long)(block0 + m_local) * DMODEL;            // piece proj is per tile id

    v8f accS = {};   // tilepos @ W_tile_top  +  Aaug @ Baug   (chained)
    v8f acc2 = {};   // piece_tile_emb @ W_piece_bot  (scaled by cnt in epilogue)
    #pragma unroll
    for (int k0 = 0; k0 < DMODEL; k0 += 4) {
        const int kk = k0 + hi * 2;
        v2f a1 = *(const v2f*)(tile_pos_emb   + rowA + kk);
        v2f a2 = *(const v2f*)(piece_tile_emb + rowB + kk);
        v2f b1, b2;
        b1[0] = tile_fc_w [(kk    ) * DMODEL + n0 + hl];
        b1[1] = tile_fc_w [(kk + 1) * DMODEL + n0 + hl];
        b2[0] = piece_fc_w[(DMODEL + kk    ) * DMODEL + n0 + hl];
        b2[1] = piece_fc_w[(DMODEL + kk + 1) * DMODEL + n0 + hl];
        accS = __builtin_amdgcn_wmma_f32_16x16x4_f32(false, a1, false, b1,
                                                     (short)0, accS, false, false);
        acc2 = __builtin_amdgcn_wmma_f32_16x16x4_f32(false, a2, false, b2,
                                                     (short)0, acc2, false, false);
    }
    // augmented K=20 GEMM chained onto accS (nation mix + biases + terrain + active)
    #pragma unroll
    for (int k0 = 0; k0 < KAUG; k0 += 4) {
        const int kk = k0 + hi * 2;
        v2f a3 = *(const v2f*)(s_afrag  + (rtile * 16 + hl) * KAUG + kk);
        v2f b3 = *(const v2f*)(s_bfragT + (n0 + hl) * KAUG + kk);
        accS = __builtin_amdgcn_wmma_f32_16x16x4_f32(false, a3, false, b3,
                                                     (short)0, accS, false, false);
    }

    // epilogue: one FMA per element
    const int d = n0 + hl;                  // this lane's output column
    #pragma unroll
    for (int v = 0; v < 8; ++v) {
        const int r_local = rtile * 16 + hi * 8 + v;   // C/D layout: VGPR v -> row v (+8 for hi lanes)
        s_state[r_local * 65 + d] = accS[v] + s_cnt[r_local] * acc2[v];
    }
    __syncthreads();

    // policy logits: one thread per row, conflict-free thanks to stride-65 padding
    if (tid < 32) {
        float s = 0.0f;
        #pragma unroll 8
        for (int dd = 0; dd < DMODEL; ++dd)
            s += s_state[tid * 65 + dd] * policy_w[dd];
        out_logits[block0 + tid] = s + policy_b[0];
    }
    // pooled partial column sums -> global accumulator
    if (tid < DMODEL) {
        float cs = 0.0f;
        #pragma unroll 8
        for (int r = 0; r < 32; ++r) cs += s_state[r * 65 + tid];
        atomicAdd(&pooled_ws[tid], cs);
    }
}

// ---------------- value head + end_turn logit ----------------
__global__ void k_finalize(const float* __restrict__ pooled_ws,
                           const float* __restrict__ value_w1,
                           const float* __restrict__ value_b1,
                           const float* __restrict__ value_w2,
                           const float* __restrict__ value_b2,
                           const float* __restrict__ end_turn,
                           float* __restrict__ d_out) {
    __shared__ float s_red[DMODEL];
    const int j = threadIdx.x;
    const float inv = 1.0f / (float)T_TILES;
    float h = value_b1[j];
    for (int d = 0; d < DMODEL; ++d)
        h += (pooled_ws[d] * inv) * value_w1[d * DMODEL + j];
    h = fmaxf(h, 0.0f);
    s_red[j] = h * value_w2[j];
    __syncthreads();
    for (int s = DMODEL / 2; s > 0; s >>= 1) {
        if (j < s) s_red[j] += s_red[j + s];
        __syncthreads();
    }
    if (j == 0) {
        d_out[T_TILES]     = end_turn[0];
        d_out[T_TILES + 1] = s_red[0] + value_b2[0];
    }
}

extern "C" void kernel_launch(void* const* d_in, const int* in_sizes, int n_in,
                              void* d_out, int out_size, void* d_ws, size_t ws_size,
                              hipStream_t stream) {
    const int*   tile_idxs       = (const int*)d_in[0];
    const int*   terrain_types   = (const int*)d_in[1];
    const int*   nation_idxs     = (const int*)d_in[2];
    const int*   piece_tile_idxs = (const int*)d_in[3];
    const int*   active_nation   = (const int*)d_in[4];
    const float* tile_pos_emb    = (const float*)d_in[5];
    const float* terrain_emb     = (const float*)d_in[6];
    const float* nation_emb      = (const float*)d_in[7];
    const float* piece_tile_emb  = (const float*)d_in[8];
    const float* tile_fc_w       = (const float*)d_in[9];
    const float* tile_fc_b       = (const float*)d_in[10];
    const float* piece_fc_w      = (const float*)d_in[11];
    const float* piece_fc_b      = (const float*)d_in[12];
    const float* tile_policy_w   = (const float*)d_in[13];
    const float* tile_policy_b   = (const float*)d_in[14];
    const float* end_turn        = (const float*)d_in[15];
    const float* value_w1        = (const float*)d_in[16];
    const float* value_b1        = (const float*)d_in[17];
    const float* value_w2        = (const float*)d_in[18];
    const float* value_b2        = (const float*)d_in[19];
    float* out = (float*)d_out;

    // workspace layout
    char* ws = (char*)d_ws;
    unsigned int* natcnt   = (unsigned int*)ws;                                  // T*16 u32 (12.8 MB)
    float*        natproj  = (float*)(ws + (size_t)T_TILES * NNATIONS * 4);      // 16*64
    float*        terrproj = natproj + NNATIONS * DMODEL;                        // 2*64
    float*        act      = terrproj + 2 * DMODEL;                              // 64
    float*        pooled   = act + DMODEL;                                       // 64

    {
        const int n = T_TILES * NNATIONS;
        k_zero<<<(n + 255) / 256, 256, 0, stream>>>(natcnt, pooled);
    }
    k_precompute<<<1, 256, 0, stream>>>(nation_emb, terrain_emb, piece_fc_w, tile_fc_w,
                                        active_nation, natproj, terrproj, act);
    k_hist<<<(NPIECES + 255) / 256, 256, 0, stream>>>(nation_idxs, piece_tile_idxs, natcnt);
    k_main<<<T_TILES / 32, 256, 0, stream>>>(tile_idxs, terrain_types,
                                             tile_pos_emb, piece_tile_emb,
                                             tile_fc_w, tile_fc_b, piece_fc_w, piece_fc_b,
                                             natcnt, natproj, terrproj, act,
                                             tile_policy_w, tile_policy_b, out, pooled);
    k_finalize<<<1, DMODEL, 0, stream>>>(pooled, value_w1, value_b1, value_w2, value_b2,
                                         end_turn, out);
}